// OverlapPatchMerging_10668698764008
// MI455X (gfx1250) — compile-verified
//
#include <hip/hip_runtime.h>

typedef __attribute__((ext_vector_type(2))) float v2f;
typedef __attribute__((ext_vector_type(8))) float v8f;

#define B_       2
#define CIN      3
#define COUT     64
#define HIN      721
#define WIN      1440
#define HOUT     361
#define WOUT     720
#define KTAP     9
#define NNZ      96
#define LN_EPS   1e-5f

#define ASTRIDE  98                 // 96 + 2 pad (keeps float2 loads 8B aligned, spreads banks)
#define NWAVES   6
#define NTHREADS (NWAVES * 32)      // 192 = 2 * NNZ (exploited in the A-build loop)
#define PLANE    (HIN * WIN)        // 1,038,240 elements per (b,c) plane
#define NTOT     (B_ * WOUT)        // 1440 columns per h

__global__ __launch_bounds__(NTHREADS)
void opm_gather_gemm_ln_kernel(const float* __restrict__ x,
                               const float* __restrict__ weight,
                               const float* __restrict__ psi,
                               const int*   __restrict__ idx_hi,
                               const int*   __restrict__ idx_dwi,
                               const float* __restrict__ lnw,
                               const float* __restrict__ lnb,
                               float*       __restrict__ out)
{
    __shared__ float psi_lds[KTAP * NNZ];        // 864 f  : psi[:, h, :]
    __shared__ float w_lds[COUT * CIN * KTAP];   // 1728 f : full weight
    __shared__ int2  base_lds[NNZ];              // (row*WIN, dwi) per e
    __shared__ float A_lds[COUT * ASTRIDE];      // 6272 f : A chunk, row-major [o][e]
    __shared__ float lnw_lds[COUT];
    __shared__ float lnb_lds[COUT];

    const int tid  = threadIdx.x;
    const int h    = blockIdx.y;
    const int lane = tid & 31;
    const int wid  = tid >> 5;
    const int l16  = lane & 15;
    const int half = lane >> 4;                  // 0: lanes 0-15, 1: lanes 16-31

    // ---- stage per-h constants into LDS ----
    for (int i = tid; i < KTAP * NNZ; i += NTHREADS) {
        int k9 = i / NNZ, e = i - k9 * NNZ;
        psi_lds[i] = psi[(k9 * HOUT + h) * NNZ + e];
    }
    for (int i = tid; i < COUT * CIN * KTAP; i += NTHREADS)
        w_lds[i] = weight[i];
    for (int i = tid; i < NNZ; i += NTHREADS)
        base_lds[i] = make_int2(idx_hi[h * NNZ + i] * WIN, idx_dwi[h * NNZ + i]);
    for (int i = tid; i < COUT; i += NTHREADS) {
        lnw_lds[i] = lnw[i];
        lnb_lds[i] = lnb[i];
    }

    // A-build ownership: thread owns column e, rows o = o0, o0+2, ... (32 rows)
    const int eown = tid & (NNZ - 1);            // tid % 96 (NNZ=96 < 128; tid<192 -> use real mod)
    const int eA_own = tid - (tid >= NNZ ? NNZ : 0);   // tid % 96 without div
    const int o0     = (tid >= NNZ) ? 1 : 0;           // tid / 96
    (void)eown;

    // this wave's 16-column strip: n = b*720 + w  (96*15 = 1440, tiles never straddle b)
    const int n    = blockIdx.x * (NWAVES * 16) + wid * 16 + l16;
    const int b    = n / WOUT;
    const int w    = n - b * WOUT;
    const int wcol = w << 1;                     // stride-2 horizontal sampling

    v8f acc[4];
    #pragma unroll
    for (int mt = 0; mt < 4; ++mt) acc[mt] = (v8f){0.f,0.f,0.f,0.f,0.f,0.f,0.f,0.f};

    // ---- K loop in chunks of 96 (one chunk per input channel c) ----
    for (int c = 0; c < CIN; ++c) {
        __syncthreads();   // previous chunk's GEMM done before overwriting A_lds

        // A_c[o][e] = sum_k9 weight[o,c,k9] * psi[k9,h,e]
        // Hoist the 9 psi values for this thread's column e into registers,
        // then sweep o with LDS-broadcast weight reads (no division, no conflicts).
        {
            float p[KTAP];
            #pragma unroll
            for (int k9 = 0; k9 < KTAP; ++k9)
                p[k9] = psi_lds[k9 * NNZ + eA_own];
            for (int o = o0; o < COUT; o += 2) {
                const float* wp = &w_lds[o * (CIN * KTAP) + c * KTAP];
                float s = 0.f;
                #pragma unroll
                for (int k9 = 0; k9 < KTAP; ++k9)
                    s += wp[k9] * p[k9];
                A_lds[o * ASTRIDE + eA_own] = s;
            }
        }
        __syncthreads();

        const int plane = (b * CIN + c) * PLANE;

        #pragma unroll 2
        for (int e0 = 0; e0 < NNZ; e0 += 4) {
            // per-lane K indices for this 16x16x4 step: eA, eA+1
            const int eA = e0 + (half << 1);

            // B fragment: gather two fp32 elements from x (served from L2)
            const int2 r0 = base_lds[eA];
            const int2 r1 = base_lds[eA + 1];
            int c0 = r0.y + wcol; c0 -= (c0 >= WIN) ? WIN : 0;
            int c1 = r1.y + wcol; c1 -= (c1 >= WIN) ? WIN : 0;
            v2f bf;
            bf.x = x[plane + r0.x + c0];
            bf.y = x[plane + r1.x + c1];

            // A fragments for the 4 M-tiles (float2 LDS loads, 8B aligned)
            v2f a0 = *(const v2f*)&A_lds[( 0 + l16) * ASTRIDE + eA];
            v2f a1 = *(const v2f*)&A_lds[(16 + l16) * ASTRIDE + eA];
            v2f a2 = *(const v2f*)&A_lds[(32 + l16) * ASTRIDE + eA];
            v2f a3 = *(const v2f*)&A_lds[(48 + l16) * ASTRIDE + eA];

            acc[0] = __builtin_amdgcn_wmma_f32_16x16x4_f32(false, a0, false, bf, (short)0, acc[0], false, false);
            acc[1] = __builtin_amdgcn_wmma_f32_16x16x4_f32(false, a1, false, bf, (short)0, acc[1], false, false);
            acc[2] = __builtin_amdgcn_wmma_f32_16x16x4_f32(false, a2, false, bf, (short)0, acc[2], false, false);
            acc[3] = __builtin_amdgcn_wmma_f32_16x16x4_f32(false, a3, false, bf, (short)0, acc[3], false, false);
        }
    }

    // ---- LayerNorm over the 64 channels of this lane's column ----
    // C/D layout: VGPR r holds (M = mt*16 + r + 8*half, N = l16)
    float s = 0.f, ss = 0.f;
    #pragma unroll
    for (int mt = 0; mt < 4; ++mt) {
        #pragma unroll
        for (int r = 0; r < 8; ++r) {
            float v = acc[mt][r];
            s  += v;
            ss += v * v;
        }
    }
    // combine with half-wave partner (lane ^ 16) -> full 64-channel sums
    s  += __shfl_xor(s, 16, 32);
    ss += __shfl_xor(ss, 16, 32);
    const float mean = s * (1.f / COUT);
    const float var  = ss * (1.f / COUT) - mean * mean;
    const float rstd = rsqrtf(var + LN_EPS);

    float* outp = out + (size_t)b * COUT * HOUT * WOUT + (size_t)h * WOUT + w;
    #pragma unroll
    for (int mt = 0; mt < 4; ++mt) {
        #pragma unroll
        for (int r = 0; r < 8; ++r) {
            const int o = mt * 16 + r + (half << 3);
            const float v = (acc[mt][r] - mean) * rstd * lnw_lds[o] + lnb_lds[o];
            outp[(size_t)o * (HOUT * WOUT)] = v;
        }
    }
}

extern "C" void kernel_launch(void* const* d_in, const int* in_sizes, int n_in,
                              void* d_out, int out_size, void* d_ws, size_t ws_size,
                              hipStream_t stream) {
    (void)in_sizes; (void)n_in; (void)d_ws; (void)ws_size; (void)out_size;
    const float* x       = (const float*)d_in[0];
    const float* weight  = (const float*)d_in[1];
    const float* psi     = (const float*)d_in[2];
    const int*   idx_hi  = (const int*)d_in[3];
    const int*   idx_dwi = (const int*)d_in[4];
    const float* lnw     = (const float*)d_in[5];
    const float* lnb     = (const float*)d_in[6];
    float*       out     = (float*)d_out;

    dim3 grid(NTOT / (NWAVES * 16), HOUT, 1);   // (15, 361)
    dim3 block(NTHREADS, 1, 1);                 // 192 threads = 6 waves
    opm_gather_gemm_ln_kernel<<<grid, block, 0, stream>>>(
        x, weight, psi, idx_hi, idx_dwi, lnw, lnb, out);
}